// CenterLoss_6837587935267
// MI455X (gfx1250) — compile-verified
//
#include <hip/hip_runtime.h>

typedef __attribute__((ext_vector_type(2))) float v2f;
typedef __attribute__((ext_vector_type(8))) float v8f;

namespace {
constexpr int kN = 8192;
constexpr int kD = 128;
constexpr int kWaves = 4;                                   // waves per block
constexpr int kBlockThreads = kWaves * 32;                  // 128 (wave32)
constexpr int kSamplesPerWave = 16;                         // one WMMA tile
constexpr int kSamplesPerBlock = kWaves * kSamplesPerWave;  // 64
constexpr int kBlocks = kN / kSamplesPerBlock;              // 128
}

// Computes, per wave, the 16 values ||x_i||^2 + ||c_{label_i}||^2 - 2 x_i.c_{label_i}
// using chained V_WMMA_F32_16X16X4_F32 (diag of X_tile @ Cgather_tile^T),
// clamps to [1e-12, 1e12] and accumulates a per-block partial sum.
__global__ __launch_bounds__(kBlockThreads)
void center_loss_partial_kernel(const float* __restrict__ X,
                                const float* __restrict__ C,
                                const int* __restrict__ labels,
                                float* __restrict__ partial) {
  __shared__ float diag_lds[kWaves][8][32];
  __shared__ float wave_sum[kWaves];

  const int lane = threadIdx.x & 31;
  const int wave = threadIdx.x >> 5;
  const int m    = lane & 15;  // tile row (A) == tile col (B) for this lane
  const int hi   = lane >> 4;  // K-half selector (ISA 16x4 f32 A/B layout)

  const int rowBase = blockIdx.x * kSamplesPerBlock + wave * kSamplesPerWave;
  const int row  = rowBase + m;
  const int cidx = labels[row];  // gathered center for sample 'row'

  const v2f* __restrict__ xrow = (const v2f*)(X + (size_t)row  * kD);
  const v2f* __restrict__ crow = (const v2f*)(C + (size_t)cidx * kD);

  v8f acc = {0.f, 0.f, 0.f, 0.f, 0.f, 0.f, 0.f, 0.f};
  float xsq = 0.f, csq = 0.f;

  // K = 128 in 32 steps of K=4. Lane layout:
  //   A: lanes 0-15 -> (M=m, K=4t..4t+1), lanes 16-31 -> (M=m, K=4t+2..4t+3)
  //   B mirrors with N=m. float2 index = 2t + hi.
#pragma unroll
  for (int t = 0; t < kD / 4; ++t) {
    v2f a = xrow[2 * t + hi];
    v2f b = crow[2 * t + hi];
    acc = __builtin_amdgcn_wmma_f32_16x16x4_f32(
        /*neg_a=*/false, a, /*neg_b=*/false, b,
        /*c_mod=*/(short)0, acc, /*reuse_a=*/false, /*reuse_b=*/false);
    xsq = fmaf(a[0], a[0], fmaf(a[1], a[1], xsq));
    csq = fmaf(b[0], b[0], fmaf(b[1], b[1], csq));
  }

  // Each lane holds half of its row's sum-of-squares; partner is lane^16.
  xsq += __shfl_xor(xsq, 16, 32);
  csq += __shfl_xor(csq, 16, 32);

  // Extract diagonal of accumulator: VGPR j holds (M=j, N=lane) for lanes 0-15
  // and (M=j+8, N=lane-16) for lanes 16-31.
#pragma unroll
  for (int j = 0; j < 8; ++j) diag_lds[wave][j][lane] = acc[j];
  __syncthreads();

  float contrib = 0.f;
  if (lane < 16) {
    const int i = lane;
    const float d = (i < 8) ? diag_lds[wave][i][i]
                            : diag_lds[wave][i - 8][i + 16];
    float sel = xsq + csq - 2.0f * d;
    sel = fminf(fmaxf(sel, 1e-12f), 1e12f);
    contrib = sel;
  }

  // Wave reduction (hi lanes contribute 0).
#pragma unroll
  for (int off = 16; off > 0; off >>= 1)
    contrib += __shfl_xor(contrib, off, 32);

  if (lane == 0) wave_sum[wave] = contrib;
  __syncthreads();

  if (threadIdx.x == 0) {
    float s = 0.f;
#pragma unroll
    for (int w = 0; w < kWaves; ++w) s += wave_sum[w];
    partial[blockIdx.x] = s;
  }
}

__global__ __launch_bounds__(128)
void center_loss_reduce_kernel(const float* __restrict__ partial,
                               float* __restrict__ out) {
  __shared__ float lds[4];
  float v = partial[threadIdx.x];  // exactly kBlocks == 128 partials
#pragma unroll
  for (int off = 16; off > 0; off >>= 1) v += __shfl_xor(v, off, 32);
  const int lane = threadIdx.x & 31;
  const int wave = threadIdx.x >> 5;
  if (lane == 0) lds[wave] = v;
  __syncthreads();
  if (threadIdx.x == 0) {
    out[0] = (lds[0] + lds[1] + lds[2] + lds[3]) / (float)kN;
  }
}

extern "C" void kernel_launch(void* const* d_in, const int* in_sizes, int n_in,
                              void* d_out, int out_size, void* d_ws, size_t ws_size,
                              hipStream_t stream) {
  const float* X      = (const float*)d_in[0];   // [8192, 128] f32
  const float* C      = (const float*)d_in[1];   // [32000, 128] f32
  const int*   labels = (const int*)d_in[2];     // [8192] (harness int convention)
  float* out     = (float*)d_out;                // scalar f32 mean
  float* partial = (float*)d_ws;                 // 128 floats of scratch

  center_loss_partial_kernel<<<kBlocks, kBlockThreads, 0, stream>>>(X, C, labels, partial);
  center_loss_reduce_kernel<<<1, 128, 0, stream>>>(partial, out);
}